// EquivariantSparseAttention_19396072309329
// MI455X (gfx1250) — compile-verified
//
#include <hip/hip_runtime.h>

typedef __attribute__((ext_vector_type(2))) float v2f;
typedef __attribute__((ext_vector_type(8))) float v8f;

#define N_NODES 10000
#define KNB 16
#define EDGE_DIM 32
#define KTOT 1040          // 64*16 product terms + 16 bias rows
#define NOUT 48
#define H_STRIDE 65
#define T_STRIDE 18

// B'[k][n]: k<1024 -> W2[(n*16 + (k&15))*64 + (k>>4)]; k>=1024 -> b2[n*16 + k-1024]
__global__ void build_bprime(const float* __restrict__ W2, const float* __restrict__ b2,
                             float* __restrict__ Bp) {
  int idx = blockIdx.x * 256 + threadIdx.x;
  if (idx >= KTOT * NOUT) return;
  int k = idx / NOUT, n = idx - k * NOUT;
  float v;
  if (k < 1024) {
    int hh = k >> 4, j = k & 15;
    v = W2[(n * 16 + j) * 64 + hh];
  } else {
    v = b2[n * 16 + (k - 1024)];
  }
  Bp[idx] = v;
}

__launch_bounds__(256)
__global__ void equi_attn_kernel(const float* __restrict__ basis1,
                                 const float* __restrict__ basis2,
                                 const float* __restrict__ ef,
                                 const float* __restrict__ f,
                                 const float* __restrict__ W1,
                                 const float* __restrict__ b1,
                                 const int* __restrict__ nbr,
                                 const float* __restrict__ Bp,
                                 float* __restrict__ out) {
  extern __shared__ float smem[];
  float* sB = smem;                         // [KTOT*NOUT] = 49920 f
  float* sH = sB + KTOT * NOUT;             // 8 waves * [16*H_STRIDE]
  float* sT = sH + 8 * 16 * H_STRIDE;       // 8 waves * [16*T_STRIDE]

  const int tid  = threadIdx.x;
  const int wave = tid >> 5;
  const int lane = tid & 31;
  const int half = lane >> 4;
  const int le   = lane & 15;

  float* myH = sH + wave * 16 * H_STRIDE;
  float* myT = sT + wave * 16 * T_STRIDE;

  // stage B' into LDS once per block (reused by 8 wave-tiles)
  for (int i = tid; i < KTOT * NOUT; i += 256) sB[i] = Bp[i];

  const int e0 = blockIdx.x * 128 + wave * 16;  // this wave's 16-edge tile
  const int e  = e0 + le;                        // per-lane edge (dup across halves)

  __builtin_prefetch(ef + (size_t)e * EDGE_DIM, 0, 1);

  // ---- tmp[e,j] = einsum('md,dl', f[src], basis1[e]), j = m*2+l ----
  if (lane < 16) {
    int src = nbr[e];
    const float4* fp = (const float4*)(f + (size_t)src * 32);
    const float4* bp = (const float4*)(basis1 + (size_t)e * 8);
    float4 bla = bp[0], blb = bp[1];   // basis1[e][d][l], d-major
#pragma unroll
    for (int m = 0; m < 8; ++m) {
      float4 fr = fp[m];
      float t0 = fr.x * bla.x + fr.y * bla.z + fr.z * blb.x + fr.w * blb.z;
      float t1 = fr.x * bla.y + fr.y * bla.w + fr.z * blb.y + fr.w * blb.w;
      myT[le * T_STRIDE + 2 * m]     = t0;
      myT[le * T_STRIDE + 2 * m + 1] = t1;
    }
  }

  // ---- layer 1: H(16x64) = relu(ef_tile(16x32) @ W1^T + b1) via WMMA f32 ----
  v8f z = {0, 0, 0, 0, 0, 0, 0, 0};
  v8f c1[4] = {z, z, z, z};
#pragma unroll
  for (int s = 0; s < 8; ++s) {
    int k0 = 4 * s + 2 * half;
    float2 a = *(const float2*)(ef + (size_t)e * EDGE_DIM + k0);
    v2f A = {a.x, a.y};
#pragma unroll
    for (int nt = 0; nt < 4; ++nt) {
      float2 bw = *(const float2*)(W1 + (size_t)(nt * 16 + le) * EDGE_DIM + k0);
      v2f B = {bw.x, bw.y};
      c1[nt] = __builtin_amdgcn_wmma_f32_16x16x4_f32(false, A, false, B, (short)0,
                                                     c1[nt], false, false);
    }
  }
#pragma unroll
  for (int nt = 0; nt < 4; ++nt) {
#pragma unroll
    for (int r = 0; r < 8; ++r) {
      int hh = nt * 16 + le;
      int er = r + 8 * half;
      float v = c1[nt][r] + b1[hh];
      myH[er * H_STRIDE + hh] = fmaxf(v, 0.0f);
    }
  }
  __syncthreads();  // B' staged; H/tmp tiles visible

  // per-lane tmp registers: this lane only ever needs j in {4c+2h, 4c+2h+1}
  float2 t01[4];
#pragma unroll
  for (int q = 0; q < 4; ++q)
    t01[q] = *(const float2*)(myT + le * T_STRIDE + 4 * q + 2 * half);

  // ---- main GEMM: out48 = (h (x) tmp) @ B', K = 1040 ----
  v8f acc[3] = {z, z, z};
  for (int g = 0; g < 65; ++g) {
    float hv = 1.0f;                       // g==64: bias rows, A' = tmp directly
    if (g < 64) hv = myH[le * H_STRIDE + g];
#pragma unroll
    for (int c = 0; c < 4; ++c) {
      int k0 = 16 * g + 4 * c + 2 * half;
      v2f A = {hv * t01[c].x, hv * t01[c].y};
#pragma unroll
      for (int nt = 0; nt < 3; ++nt) {
        int n = nt * 16 + le;
        v2f B = {sB[k0 * NOUT + n], sB[(k0 + 1) * NOUT + n]};
        acc[nt] = __builtin_amdgcn_wmma_f32_16x16x4_f32(false, A, false, B, (short)0,
                                                        acc[nt], false, false);
      }
    }
  }

  // ---- out48 tile -> LDS (reuse myH region, stride 49) ----
  float* myO = myH;
#pragma unroll
  for (int nt = 0; nt < 3; ++nt)
#pragma unroll
    for (int r = 0; r < 8; ++r)
      myO[(r + 8 * half) * 49 + nt * 16 + le] = acc[nt][r];

  __syncthreads();  // all waves done reading sB -> safe to recycle for kqv

  // ---- kqv[e][m][d] = tmp2[m,0]*basis2[e,0,d] + tmp2[m,1]*basis2[e,1,d] ----
  float* sKQV = smem;  // 128 edges * 96 floats = 12288 f (fits in old B' region)
  {
    float4 b20 = *(const float4*)(basis2 + (size_t)e * 8);
    float4 b21 = *(const float4*)(basis2 + (size_t)e * 8 + 4);
    const float* Or = myO + le * 49;
    float4* dst = (float4*)(sKQV + (size_t)(wave * 16 + le) * 96);
#pragma unroll
    for (int mm = 0; mm < 12; ++mm) {   // halves split the 24 m-values
      int m = half * 12 + mm;
      float u0 = Or[2 * m], u1 = Or[2 * m + 1];
      float4 r;
      r.x = u0 * b20.x + u1 * b21.x;
      r.y = u0 * b20.y + u1 * b21.y;
      r.z = u0 * b20.z + u1 * b21.z;
      r.w = u0 * b20.w + u1 * b21.w;
      dst[m] = r;
    }
  }
  __syncthreads();

  // ---- attention: wave w handles node blockIdx*8 + w; lane = (head, hd) ----
  {
    const int h    = lane >> 3;
    const int hd   = lane & 7;
    const int mloc = 2 * h + (hd >> 2);
    const int d    = hd & 3;
    const float* base = sKQV + (size_t)wave * 16 * 96;

    float qn = 0.0f;
#pragma unroll
    for (int k = 0; k < 16; ++k)
      qn += base[k * 96 + (8 + mloc) * 4 + d];
    qn *= 0.0625f;  // mean over K

    float sc[16];
#pragma unroll
    for (int k = 0; k < 16; ++k) {
      float p = qn * base[k * 96 + mloc * 4 + d];
      p += __shfl_xor(p, 1, 32);
      p += __shfl_xor(p, 2, 32);
      p += __shfl_xor(p, 4, 32);           // reduce over head_dim (8 lanes)
      sc[k] = p * 0.35355339059327373f;    // * HEAD_DIM^-0.5
    }
    float mx = sc[0];
#pragma unroll
    for (int k = 1; k < 16; ++k) mx = fmaxf(mx, sc[k]);
    float sum = 0.0f;
#pragma unroll
    for (int k = 0; k < 16; ++k) { sc[k] = __expf(sc[k] - mx); sum += sc[k]; }
    float inv = 1.0f / sum;
    float o = 0.0f;
#pragma unroll
    for (int k = 0; k < 16; ++k)
      o += sc[k] * base[k * 96 + (16 + mloc) * 4 + d];
    o *= inv;

    int n = blockIdx.x * 8 + wave;
    out[n * 32 + lane] = o;   // (N,MULT,DIM) flat == (N,H,hd) flat
  }
}

extern "C" void kernel_launch(void* const* d_in, const int* in_sizes, int n_in,
                              void* d_out, int out_size, void* d_ws, size_t ws_size,
                              hipStream_t stream) {
  const float* basis1 = (const float*)d_in[0];
  const float* basis2 = (const float*)d_in[1];
  const float* ef     = (const float*)d_in[2];
  const float* f      = (const float*)d_in[3];
  const float* W1     = (const float*)d_in[4];
  const float* b1     = (const float*)d_in[5];
  const float* W2     = (const float*)d_in[6];
  const float* b2     = (const float*)d_in[7];
  const int*   nbr    = (const int*)d_in[8];
  float* out = (float*)d_out;
  float* Bp  = (float*)d_ws;   // 1040*48 floats = 199,680 bytes

  build_bprime<<<(KTOT * NOUT + 255) / 256, 256, 0, stream>>>(W2, b2, Bp);

  size_t lds_bytes = (size_t)(KTOT * NOUT + 8 * 16 * H_STRIDE + 8 * 16 * T_STRIDE) * 4;
  int nblocks = N_NODES / 8;  // 1250 blocks * 128 edges = 160000 edges exactly
  equi_attn_kernel<<<nblocks, 256, lds_bytes, stream>>>(basis1, basis2, ef, f, W1, b1,
                                                        nbr, Bp, out);
}